// SADecompLayer_644245094889
// MI455X (gfx1250) — compile-verified
//
#include <hip/hip_runtime.h>

// SADecomp: out_block = 0.125 * Q (w + w^T) Q per 8x8 block, Q = I + J.
// Per 16x16 tile (2x2 blocks) on one wave32:
//   stage tile -> LDS via GLOBAL_LOAD_ASYNC_TO_LDS_B128 (CDNA5 async path)
//   X = T + blockwise_transpose(T)      (LDS gather, A-operand layout)
//   P = X * Q2   via 4x v_wmma_f32_16x16x4_f32  (Q2 = diag(Q,Q), B operand)
//   out = 0.125 * (P + flipud_per_block(P))     (VGPR index reversal in D layout)

typedef __attribute__((ext_vector_type(2))) float v2f;
typedef __attribute__((ext_vector_type(8))) float v8f;

#define HW 1024
#define LDS_PITCH 20   // floats; 80B row pitch: 16B-aligned quads, conflict-free banking

__global__ __launch_bounds__(256) void sadecomp_d4_kernel(const float* __restrict__ in,
                                                          float* __restrict__ out) {
    // 8 waves per WG, one 16x16 tile per wave; wave-private LDS slab.
    __shared__ __align__(16) float lds[8][16 * LDS_PITCH];   // 10240 bytes

    const int lane = threadIdx.x & 31;
    const int wid  = threadIdx.x >> 5;

    const int tile = blockIdx.x * 8 + wid;        // 0 .. 262143
    const int tc = tile & 63;                     // tile col (16-wide)
    const int tr = (tile >> 6) & 63;              // tile row (16-tall)
    const int ch = tile >> 12;                    // channel 0..63

    const size_t base = ((size_t)ch * HW + (size_t)tr * 16) * HW + (size_t)tc * 16;
    const float* __restrict__ src = in + base;
    float* __restrict__ dst = out + base;
    float* sm = lds[wid];

    // ---- Stage 16x16 f32 tile straight into LDS: 2x global_load_async_to_lds_b128
    // per lane (16B each). The LDS VDST operand is the per-lane byte offset within
    // the workgroup LDS allocation == low 32 bits of the generic pointer. Deriving
    // it from the pointer (ptrtoint) CAPTURES `lds`, so the "memory"-clobbering asm
    // is assumed to write it: the ds_loads below can't be folded away or hoisted.
    {
        const int f0 = lane;          // float4 slot 0..31
        const int f1 = lane + 32;     // float4 slot 32..63
        const int r0 = f0 >> 2, q0 = f0 & 3;
        const int r1 = f1 >> 2, q1 = f1 & 3;
        const unsigned lds0 = (unsigned)(uintptr_t)(sm + r0 * LDS_PITCH + q0 * 4);
        const unsigned lds1 = (unsigned)(uintptr_t)(sm + r1 * LDS_PITCH + q1 * 4);
        const unsigned long long g0 = (unsigned long long)(src + (size_t)r0 * HW + q0 * 4);
        const unsigned long long g1 = (unsigned long long)(src + (size_t)r1 * HW + q1 * 4);
        asm volatile("global_load_async_to_lds_b128 %0, %1, off"
                     :: "v"(lds0), "v"(g0) : "memory");
        asm volatile("global_load_async_to_lds_b128 %0, %1, off"
                     :: "v"(lds1), "v"(g1) : "memory");
        // Wave-private LDS slab: wave-local wait on ASYNCcnt is sufficient.
        asm volatile("s_wait_asynccnt 0" ::: "memory");
    }
    __syncthreads();

    const int m    = lane & 15;   // A row / B,C,D column this lane owns
    const int half = lane >> 4;   // K/M half selector

    // ---- Gather A operand chunks: X[m][j] = T[m][j] + T[(m&8)|(j&7)][(j&8)|(m&7)]
    v2f a[4];
#pragma unroll
    for (int c = 0; c < 4; ++c) {
#pragma unroll
        for (int k = 0; k < 2; ++k) {
            const int j  = 4 * c + 2 * half + k;
            const int r2 = (m & 8) | (j & 7);     // blockwise-transposed row
            const int c2 = (j & 8) | (m & 7);     // blockwise-transposed col
            a[c][k] = sm[m * LDS_PITCH + j] + sm[r2 * LDS_PITCH + c2];
        }
    }

    // ---- B operand chunks: rows of Q2 = diag(I+J, I+J). Row r of Q2 has ones at
    // n==r and n==rf; test lane's column bit in the 2-bit row mask.
    v2f b[4];
#pragma unroll
    for (int c = 0; c < 4; ++c) {
#pragma unroll
        for (int k = 0; k < 2; ++k) {
            const int r  = 4 * c + 2 * half + k;          // K row within 16
            const int rf = (r & 8) | (7 - (r & 7));       // per-block antidiagonal
            const unsigned rowmask = (1u << r) | (1u << rf);
            b[c][k] = (float)((rowmask >> m) & 1u);
        }
    }

    // ---- P = X * Q2  (four K=4 slices accumulated)
    v8f acc = {};
#pragma unroll
    for (int c = 0; c < 4; ++c) {
        acc = __builtin_amdgcn_wmma_f32_16x16x4_f32(
            /*neg_a=*/false, a[c], /*neg_b=*/false, b[c],
            /*c_mod=*/(short)0, acc, /*reuse_a=*/false, /*reuse_b=*/false);
    }

    // ---- out = 0.125 * (P + flipud_per_block(P)); in D layout the per-block row
    // flip is exactly acc[r] <-> acc[7-r] for both lane halves. Each store writes
    // two contiguous 64B row segments per wave (coalesced).
#pragma unroll
    for (int r = 0; r < 8; ++r) {
        const float o = 0.125f * (acc[r] + acc[7 - r]);
        dst[(size_t)(r + 8 * half) * HW + m] = o;
    }
}

extern "C" void kernel_launch(void* const* d_in, const int* in_sizes, int n_in,
                              void* d_out, int out_size, void* d_ws, size_t ws_size,
                              hipStream_t stream) {
    (void)in_sizes; (void)n_in; (void)out_size; (void)d_ws; (void)ws_size;
    const float* in = (const float*)d_in[0];
    float* out = (float*)d_out;
    // 64 ch * 64 * 64 tiles = 262144 tiles, 8 tiles (waves) per 256-thread WG
    dim3 grid(262144 / 8), block(256);
    hipLaunchKernelGGL(sadecomp_d4_kernel, grid, block, 0, stream, in, out);
}